// RateDistortionLoss_36782099923388
// MI455X (gfx1250) — compile-verified
//
#include <hip/hip_runtime.h>

typedef __attribute__((ext_vector_type(2))) float v2f;
typedef __attribute__((ext_vector_type(8))) float v8f;

#define B_CONST      4
#define NPTS         8192
#define WG_THREADS   256
#define WAVES_PER_WG 8
#define ROWS_PER_WG  (WAVES_PER_WG * 16)   // 128 rows per workgroup
#define NTILES       (NPTS / 16)           // 512 column tiles

// One launch covers both chamfer directions (blockIdx.z): direction 1 simply
// swaps pred/target roles, so only row-mins are ever needed -> no col atomics.
__global__ __launch_bounds__(WG_THREADS)
void chamfer_rowmin_kernel(const float* __restrict__ pred,
                           const float* __restrict__ targ,
                           float* __restrict__ partials)
{
    extern __shared__ float ldsB[];  // NPTS * 4 floats = 128 KB, staged targets

    const int b   = blockIdx.y;
    const int dir = blockIdx.z;
    const float* P = dir ? targ : pred;   // "row" point set
    const float* Q = dir ? pred : targ;   // "column" point set
    const float* Pb = P + (size_t)b * NPTS * 3;
    const float* Qb = Q + (size_t)b * NPTS * 3;

    const int tid = threadIdx.x;

    // Phase 1: stage Q into LDS as (-2q0, -2q1, -2q2, |q|^2) per point.
    for (int i = tid; i < NPTS; i += WG_THREADS) {
        float q0 = Qb[i * 3 + 0];
        float q1 = Qb[i * 3 + 1];
        float q2 = Qb[i * 3 + 2];
        float4 v = make_float4(-2.0f * q0, -2.0f * q1, -2.0f * q2,
                               q0 * q0 + q1 * q1 + q2 * q2);
        *(float4*)(&ldsB[i * 4]) = v;   // ds_store_b128, 16B aligned
    }
    __syncthreads();

    const int wave  = tid >> 5;
    const int lane  = tid & 31;
    const int half  = lane >> 4;   // 0: lanes 0-15 (K0,K1) | 1: lanes 16-31 (K2,K3)
    const int r16   = lane & 15;
    const int nbase = blockIdx.x * ROWS_PER_WG + wave * 16;

    // A operand (16x4 f32): row = nbase + r16; K = (x0,x1,x2,1)
    v2f a;
    {
        const float* pr = Pb + (size_t)(nbase + r16) * 3;
        if (half == 0) { a.x = pr[0]; a.y = pr[1]; }
        else           { a.x = pr[2]; a.y = 1.0f;  }
    }

    // C operand (16x16 f32 acc layout): VGPR r, this half -> row nbase+8*half+r,
    // value |p_row|^2 broadcast across the 16 columns.
    v8f c;
    #pragma unroll
    for (int r = 0; r < 8; ++r) {
        const float* pr = Pb + (size_t)(nbase + half * 8 + r) * 3;
        c[r] = pr[0] * pr[0] + pr[1] * pr[1] + pr[2] * pr[2];
    }

    v8f rmin;
    #pragma unroll
    for (int r = 0; r < 8; ++r) rmin[r] = 3.4e38f;

    // Per lane: B operand slice lives at element (m*4 + 2*half), m = tile*16 + r16.
    // ds_load_b64 per lane per tile; banks 4k/4k+1 (half 0) vs 4k+2/4k+3 (half 1)
    // -> all 64 LDS banks hit exactly once, conflict-free.
    const float* bp = &ldsB[r16 * 4 + half * 2];

    #pragma unroll 4
    for (int mt = 0; mt < NTILES; ++mt) {
        v2f bv = *(const v2f*)(bp + mt * 64);
        // D = A x B + C  ==  full squared-distance 16x16 tile
        v8f d = __builtin_amdgcn_wmma_f32_16x16x4_f32(
            /*neg_a=*/false, a, /*neg_b=*/false, bv,
            /*c_mod=*/(short)0, c, /*reuse_a=*/false, /*reuse_b=*/false);
        #pragma unroll
        for (int r = 0; r < 8; ++r) rmin[r] = fminf(rmin[r], d[r]);
    }

    // Min across the 16 lanes of each half (columns), then sum the 8 rows.
    float s = 0.0f;
    #pragma unroll
    for (int r = 0; r < 8; ++r) {
        float v = rmin[r];
        v = fminf(v, __shfl_xor(v, 1, 32));
        v = fminf(v, __shfl_xor(v, 2, 32));
        v = fminf(v, __shfl_xor(v, 4, 32));
        v = fminf(v, __shfl_xor(v, 8, 32));
        s += v;
    }
    s += __shfl_xor(s, 16, 32);   // combine rows 0-7 (lanes<16) with rows 8-15

    if (lane == 0) {
        int idx = ((dir * B_CONST + b) * gridDim.x + blockIdx.x) * WAVES_PER_WG + wave;
        partials[idx] = s;        // deterministic: no atomics
    }
}

__global__ __launch_bounds__(256)
void chamfer_finalize_kernel(const float* __restrict__ partials,
                             const float* __restrict__ fbpp,
                             float* __restrict__ out, int nPerDir)
{
    __shared__ float red[256];
    const int tid = threadIdx.x;
    float s0 = 0.0f, s1 = 0.0f;
    for (int i = tid; i < nPerDir; i += 256) {
        s0 += partials[i];
        s1 += partials[nPerDir + i];
    }
    red[tid] = s0; __syncthreads();
    for (int k = 128; k > 0; k >>= 1) {
        if (tid < k) red[tid] += red[tid + k];
        __syncthreads();
    }
    float S0 = red[0];
    __syncthreads();
    red[tid] = s1; __syncthreads();
    for (int k = 128; k > 0; k >>= 1) {
        if (tid < k) red[tid] += red[tid + k];
        __syncthreads();
    }
    if (tid == 0) {
        float S1   = red[0];
        float inv  = 1.0f / (float)(B_CONST * NPTS);
        float dist = S0 * inv + S1 * inv;
        float rate = fbpp[0];
        out[0] = dist + rate;  // loss (LAMBDA = 1)
        out[1] = dist;
        out[2] = rate;
    }
}

extern "C" void kernel_launch(void* const* d_in, const int* in_sizes, int n_in,
                              void* d_out, int out_size, void* d_ws, size_t ws_size,
                              hipStream_t stream)
{
    const float* pred = (const float*)d_in[0];   // (4, 8192, 3) f32
    const float* targ = (const float*)d_in[1];   // (4, 8192, 3) f32
    const float* fbpp = (const float*)d_in[2];   // (1,) f32
    float* out      = (float*)d_out;             // (loss, dist, rate)
    float* partials = (float*)d_ws;              // 2*4*64*8 = 4096 floats

    dim3 grid(NPTS / ROWS_PER_WG, B_CONST, 2);   // (64, 4, 2)
    size_t ldsBytes = (size_t)NPTS * 4 * sizeof(float);  // 128 KB dynamic LDS
    hipLaunchKernelGGL(chamfer_rowmin_kernel, grid, dim3(WG_THREADS), ldsBytes,
                       stream, pred, targ, partials);

    int nPerDir = (NPTS / ROWS_PER_WG) * B_CONST * WAVES_PER_WG;  // 2048
    hipLaunchKernelGGL(chamfer_finalize_kernel, dim3(1), dim3(256), 0, stream,
                       partials, fbpp, out, nPerDir);
}